// SteerableFirstLayerPerSample_14671608283208
// MI455X (gfx1250) — compile-verified
//
#include <hip/hip_runtime.h>
#include <hip/hip_bf16.h>
#include <stdint.h>

// ---- problem constants (from reference) ----
#define FS_F      50000000.0f
#define KW        31          // taps
#define OCN       64          // out channels
#define ICN       2           // in channels (I/Q)
#define LN        16384       // signal length
#define BN        128         // batch
#define PI_F      3.14159265358979323846f

#define NCOLS     512         // output columns per block (conv kernel)
#define XROW      (NCOLS + 32) // padded staged row length per channel

typedef __attribute__((ext_vector_type(2))) float v2f;
typedef __attribute__((ext_vector_type(8))) float v8f;

// =====================================================================
// Kernel 1: per-sample weight synthesis.
// One block per batch sample. Writes W[b][o][c][k] (f32) into workspace.
// =====================================================================
__global__ __launch_bounds__(256) void build_weights_kernel(
    const float* __restrict__ s,      // (B,5): cfo, scale, gain, shift, chirp
    const int*   __restrict__ z,      // (B,5) activation flags
    const float* __restrict__ base_w, // (OC, IC, 1, KW)
    float*       __restrict__ Wout)   // (B, OC, IC, KW)
{
    __shared__ float Msm[KW][KW];     // per-batch resampling matrix

    const int b   = blockIdx.x;
    const int tid = threadIdx.x;

    // --- per-sample scalars (computed redundantly per thread; cheap) ---
    float s0 = s[b*5+0], s1 = s[b*5+1], s2 = s[b*5+2], s3 = s[b*5+3], s4 = s[b*5+4];
    if (s0 != s0) s0 = 0.f;
    if (s1 != s1) s1 = 0.f;
    if (s2 != s2) s2 = 0.f;
    if (s3 != s3) s3 = 0.f;
    if (s4 != s4) s4 = 0.f;
    const bool a0 = z[b*5+0] > 0, a1 = z[b*5+1] > 0, a2 = z[b*5+2] > 0,
               a3 = z[b*5+3] > 0, a4 = z[b*5+4] > 0;

    const float alpha  = a1 ? s1 : 1.0f;         // COEF_SCALE == 1
    const float f0     = a0 ? s0 : 0.0f;         // COEF_CFO   == 1
    const float chirp  = a4 ? s4 : 0.0f;         // COEF_CHIRP == 1
    const float rho    = a2 ? s2 : 1.0f;         // COEF_GAIN  == 1
    const int   kshift = a3 ? (int)rintf(s3) : 0;

    float new_w = rintf((float)KW / fmaxf(alpha, 0.001f));
    new_w = fminf(fmaxf(new_w, 1.0f), (float)(2 * KW));

    // --- build resampling matrix M = Wu @ Wd (each row has <= 4 nnz) ---
    if (tid < KW) {
        float row[KW];
        #pragma unroll
        for (int p = 0; p < KW; ++p) row[p] = 0.f;

        const float src_u = fmaxf(((float)tid + 0.5f) * new_w / (float)KW - 0.5f, 0.0f);
        const float u0f   = fminf(fmaxf(floorf(src_u), 0.0f), new_w - 1.0f);
        const float u1f   = fminf(u0f + 1.0f, new_w - 1.0f);
        const float fu    = src_u - u0f;

        #pragma unroll
        for (int t = 0; t < 2; ++t) {
            const float nF = (t == 0) ? u0f : u1f;
            const float wn = (t == 0) ? (1.0f - fu) : fu;
            const float src_d = fmaxf((nF + 0.5f) * (float)KW / new_w - 0.5f, 0.0f);
            const float fi0   = fminf(fmaxf(floorf(src_d), 0.0f), (float)(KW - 1));
            const int   i0    = (int)fi0;
            const int   i1    = (i0 + 1 < KW) ? i0 + 1 : KW - 1;
            const float fr    = src_d - fi0;
            row[i0] += wn * (1.0f - fr);
            row[i1] += wn * fr;
        }
        #pragma unroll
        for (int p = 0; p < KW; ++p) Msm[tid][p] = row[p];
    }
    __syncthreads();

    // --- w[o,c,j] = sum_p M[j,p] base[o,c,p]; then IQ-rotate, gain, shift ---
    const float inv_fs = 1.0f / FS_F;
    for (int idx = tid; idx < OCN * KW; idx += 256) {
        const int o = idx / KW;
        const int j = idx - o * KW;             // tap index BEFORE circular shift

        float I = 0.f, Q = 0.f;
        const float* bi = base_w + (o * ICN + 0) * KW;
        const float* bq = base_w + (o * ICN + 1) * KW;
        for (int p = 0; p < KW; ++p) {
            const float m = Msm[j][p];
            I = fmaf(m, bi[p], I);
            Q = fmaf(m, bq[p], Q);
        }

        // combined CFO + chirp rotation: phi = 2*pi*f0*n/FS + pi*a*(n/FS)^2
        const float tn  = (float)j * inv_fs;
        const float phi = 2.0f * PI_F * f0 * (float)j * inv_fs + PI_F * chirp * tn * tn;
        const float cp = cosf(phi), sp = sinf(phi);
        const float I2 = rho * (I * cp - Q * sp);
        const float Q2 = rho * (I * sp + Q * cp);

        const int ko = ((j + kshift) % KW + KW) % KW;   // new position of old tap j
        Wout[((b * OCN + o) * ICN + 0) * KW + ko] = I2;
        Wout[((b * OCN + o) * ICN + 1) * KW + ko] = Q2;
    }
}

// =====================================================================
// Kernel 2: per-sample conv as implicit GEMM with V_WMMA_F32_16X16X4_F32.
//   Y[b][o][l] = sum_{c,k} W[b][o][c][k] * x[b][c][l+k-15]
// Grid: (LN/NCOLS, OCN/16, BN).  256 threads = 8 waves.
// x slice staged via async global->LDS (ASYNCcnt); output stored with TH=NT
// (537 MB streamed once; keep the 192 MB L2 for the re-read x slices).
// =====================================================================
__global__ __launch_bounds__(256) void conv_wmma_kernel(
    const float* __restrict__ x,    // (B, IC, 1, LN)
    const float* __restrict__ W,    // (B, OC, IC, KW) from workspace
    float*       __restrict__ y)    // (B, OC, 1, LN)
{
    __shared__ float Wsm[16][64];        // 16 OCs x K'(=62, zero-padded to 64)
    __shared__ float xsm[ICN][XROW];     // staged input slice (+halo)

    const int b     = blockIdx.z;
    const int oc0   = blockIdx.y * 16;
    const int nbase = blockIdx.x * NCOLS;
    const int tid   = threadIdx.x;

    // ---- stage weight tile: row r = c*KW + k; rows 62,63 are zero pad ----
    for (int t = tid; t < 16 * 64; t += 256) {
        const int m = t >> 6;
        const int r = t & 63;
        float v = 0.f;
        if (r < ICN * KW) {
            const int c = (r < KW) ? 0 : 1;
            const int k = r - c * KW;
            v = W[((b * OCN + oc0 + m) * ICN + c) * KW + k];
        }
        Wsm[m][r] = v;
    }

    // ---- stage input slice: xsm[c][j] = x[b,c, nbase + j - 15] (zero OOB) ----
    // In-range elements: async global->LDS copy (ASYNCcnt path, no VGPR bounce).
    // Out-of-range / pad elements: plain zero ds_store (disjoint addresses).
    {
        const float* xb = x + (size_t)b * ICN * LN;   // SGPR base for GVS mode
        for (int t = tid; t < ICN * XROW; t += 256) {
            const int c  = t / XROW;
            const int j  = t - c * XROW;
            const int gl = nbase + j - (KW / 2);
            if (j < NCOLS + KW - 1 && gl >= 0 && gl < LN) {
                const uint32_t lds_addr = (uint32_t)(uintptr_t)&xsm[c][j];
                const uint32_t goff     = (uint32_t)((c * LN + gl) * 4);
                asm volatile("global_load_async_to_lds_b32 %0, %1, %2"
                             :: "v"(lds_addr), "v"(goff), "s"(xb)
                             : "memory");
            } else {
                xsm[c][j] = 0.f;
            }
        }
        asm volatile("s_wait_asynccnt 0" ::: "memory");
    }
    __syncthreads();

    const int lane    = tid & 31;
    const int wave    = tid >> 5;
    const int mrow    = lane & 15;   // A-matrix row (OC within tile)
    const int half    = lane >> 4;   // half-wave (k-pair / M+8 select)
    const int ncol    = lane & 15;   // B/D column within 16-wide tile
    const int colbase = wave * 64;   // this wave's 64-column strip

    v8f acc0 = {}, acc1 = {}, acc2 = {}, acc3 = {};

    #pragma unroll
    for (int kk = 0; kk < 16; ++kk) {               // K' = 64 in steps of 4
        const int kb = (kk << 2) + (half << 1);     // this lane's k-pair base

        // A fragment (16x4 f32): lanes 0-15 hold k={kb,kb+1}, 16-31 k={kb+2,kb+3}
        v2f A;
        A.x = Wsm[mrow][kb];
        A.y = Wsm[mrow][kb + 1];

        // decode B rows kb, kb+1 -> (channel, tap); rows >= 62 are zero
        const int  r0 = kb, r1 = kb + 1;
        const int  c0 = (r0 < KW) ? 0 : 1;
        const int  k0 = r0 - c0 * KW;
        const int  c1 = (r1 < KW) ? 0 : 1;
        const int  k1 = r1 - c1 * KW;
        const bool v0 = (r0 < ICN * KW);
        const bool v1 = (r1 < ICN * KW);

        #pragma unroll
        for (int nt = 0; nt < 4; ++nt) {
            const int j = colbase + nt * 16 + ncol; // local output column
            v2f Bf;
            Bf.x = v0 ? xsm[c0][j + k0] : 0.f;
            Bf.y = v1 ? xsm[c1][j + k1] : 0.f;
            v8f& acc = (nt == 0) ? acc0 : (nt == 1) ? acc1 : (nt == 2) ? acc2 : acc3;
            // D = A x B + C  (f32, exact reference numerics)
            acc = __builtin_amdgcn_wmma_f32_16x16x4_f32(
                      /*neg_a=*/false, A, /*neg_b=*/false, Bf,
                      /*c_mod=*/(short)0, acc,
                      /*reuse_a=*/false, /*reuse_b=*/false);
        }
    }

    // ---- store: D layout => VGPR v holds M=v (lanes 0-15) / M=v+8 (16-31) ----
    // Non-temporal: output is write-once streaming traffic (>> L2 capacity).
    #pragma unroll
    for (int nt = 0; nt < 4; ++nt) {
        const v8f& acc = (nt == 0) ? acc0 : (nt == 1) ? acc1 : (nt == 2) ? acc2 : acc3;
        const int l = nbase + colbase + nt * 16 + ncol;
        #pragma unroll
        for (int v = 0; v < 8; ++v) {
            const int o = oc0 + v + half * 8;
            __builtin_nontemporal_store(acc[v], &y[(size_t)(b * OCN + o) * LN + l]);
        }
    }
}

// =====================================================================
// Host entry
// =====================================================================
extern "C" void kernel_launch(void* const* d_in, const int* in_sizes, int n_in,
                              void* d_out, int out_size, void* d_ws, size_t ws_size,
                              hipStream_t stream) {
    const float* x      = (const float*)d_in[0];  // (B, IC, 1, L) f32
    const int*   z      = (const int*)  d_in[1];  // (B, 5) i32
    const float* s      = (const float*)d_in[2];  // (B, 5) f32
    const float* base_w = (const float*)d_in[3];  // (OC, IC, 1, KW) f32
    float*       out    = (float*)d_out;          // (B, OC, 1, L) f32
    float*       Wws    = (float*)d_ws;           // B*OC*IC*KW floats (~2 MB)

    build_weights_kernel<<<dim3(BN), dim3(256), 0, stream>>>(s, z, base_w, Wws);

    conv_wmma_kernel<<<dim3(LN / NCOLS, OCN / 16, BN), dim3(256), 0, stream>>>(
        x, Wws, out);
}